// OpeningLoss2D_18124761989638
// MI455X (gfx1250) — compile-verified
//
#include <hip/hip_runtime.h>

// Problem constants (from the reference): [N=16, C=4, H=1024, W=1024] f32.
constexpr int Himg = 1024;
constexpr int Wimg = 1024;
constexpr int NIMG = 16 * 4;            // N*C independent 2D slices
constexpr int TILE = 64;                // 64x64 outputs per block
constexpr int RAW  = TILE + 2;          // 66x66 halo tile of input
constexpr int ERO  = TILE + 1;          // 65x65 erosion region
constexpr int LSTR = 68;                // padded LDS row stride (floats)
constexpr int TPB  = 256;               // 8 wave32s
constexpr int TILES_X = Wimg / TILE;    // 16
constexpr int TILES_Y = Himg / TILE;    // 16
constexpr int NBLOCKS = NIMG * TILES_X * TILES_Y;  // 16384
constexpr float INV_TOTAL = 1.0f / (float)(16 * 4 * 1024 * 1024); // exact (pow2)

// ---- CDNA5 async global->LDS path (probe-guarded) -------------------------
#if defined(__HIP_DEVICE_COMPILE__) && \
    __has_builtin(__builtin_amdgcn_global_load_async_to_lds_b32)
#define HAVE_ASYNC_LDS 1
#else
#define HAVE_ASYNC_LDS 0
#endif

#if HAVE_ASYNC_LDS
typedef __attribute__((address_space(1))) int* as1_int_p;  // global (AS1)
typedef __attribute__((address_space(3))) int* as3_int_p;  // LDS    (AS3)
#endif

__device__ __forceinline__ void stage_b32(const float* __restrict__ g,
                                          float* __restrict__ l) {
#if HAVE_ASYNC_LDS
  // emits GLOBAL_LOAD_ASYNC_TO_LDS_B32 (ASYNCcnt); per-lane LDS dest address.
  __builtin_amdgcn_global_load_async_to_lds_b32(
      (as1_int_p)(void*)g, (as3_int_p)(void*)l, /*offset=*/0, /*cpol=*/0);
#else
  *l = *g;
#endif
}

__device__ __forceinline__ void wait_async_then_barrier() {
#if HAVE_ASYNC_LDS
#if __has_builtin(__builtin_amdgcn_s_wait_asynccnt)
  __builtin_amdgcn_s_wait_asynccnt(0);
#else
  asm volatile("s_wait_asynccnt 0" ::: "memory");
#endif
#endif
  __syncthreads();
}

// ---------------------------------------------------------------------------
// Kernel 1: fused grey-opening(2x2, reflect) + squared-diff partial sums.
// One 64x64 output tile per block; LDS-staged 66x66 halo tile.
// ---------------------------------------------------------------------------
__global__ __launch_bounds__(TPB) void opening_loss_tile_kernel(
    const float* __restrict__ x, float* __restrict__ partial) {
  __shared__ float t[RAW * LSTR];   // raw halo tile  (66 rows)
  __shared__ float e[ERO * LSTR];   // erosion        (65 rows)
  __shared__ float red[TPB];

  const int tid = threadIdx.x;
  const int b   = blockIdx.x;
  const int tx  = b & (TILES_X - 1);
  const int ty  = (b >> 4) & (TILES_Y - 1);
  const int img = b >> 8;

  const float* __restrict__ base = x + (size_t)img * ((size_t)Himg * Wimg);
  const int r0 = ty * TILE;
  const int c0 = tx * TILE;

  // ---- Stage 66x66 halo with clamped (== reflect pad-1) indices ----------
  for (int idx = tid; idx < RAW * RAW; idx += TPB) {
    const int ly = idx / RAW;
    const int lx = idx - ly * RAW;
    int gy = r0 - 1 + ly;
    int gx = c0 - 1 + lx;
    gy = gy < 0 ? 0 : (gy > Himg - 1 ? Himg - 1 : gy);
    gx = gx < 0 ? 0 : (gx > Wimg - 1 ? Wimg - 1 : gx);
    stage_b32(base + (size_t)gy * Wimg + gx, &t[ly * LSTR + lx]);
  }
  wait_async_then_barrier();

  // ---- Erosion: e[i][j] = min over t[i..i+1][j..j+1], 65x65 --------------
  for (int idx = tid; idx < ERO * ERO; idx += TPB) {
    const int i = idx / ERO;
    const int j = idx - i * ERO;
    const float* r0p = &t[i * LSTR + j];
    const float* r1p = &t[(i + 1) * LSTR + j];
    const float m0 = fminf(r0p[0], r0p[1]);
    const float m1 = fminf(r1p[0], r1p[1]);
    e[i * LSTR + j] = fminf(m0, m1);
  }
  __syncthreads();

  // ---- Dilation + squared diff, 64x64 ------------------------------------
  // d[i][j] = max(e[i][j], e[i][jp], e[ip][j], e[ip][jp]),
  // ip/jp clamp ONLY at the true image border (reflect pad on bottom/right).
  float acc = 0.0f;
  for (int idx = tid; idx < TILE * TILE; idx += TPB) {
    const int i = idx >> 6;
    const int j = idx & (TILE - 1);
    const int ip = (r0 + i == Himg - 1) ? i : i + 1;
    const int jp = (c0 + j == Wimg - 1) ? j : j + 1;
    const float d00 = e[i * LSTR + j];
    const float d01 = e[i * LSTR + jp];
    const float d10 = e[ip * LSTR + j];
    const float d11 = e[ip * LSTR + jp];
    const float dmax = fmaxf(fmaxf(d00, d01), fmaxf(d10, d11));
    const float v = t[(i + 1) * LSTR + (j + 1)] - dmax;
    acc = fmaf(v, v, acc);
  }

  // ---- Block reduction ----------------------------------------------------
  red[tid] = acc;
  __syncthreads();
  for (int s = TPB / 2; s > 0; s >>= 1) {
    if (tid < s) red[tid] += red[tid + s];
    __syncthreads();
  }
  if (tid == 0) partial[b] = red[0];
}

// ---------------------------------------------------------------------------
// Kernel 2: deterministic reduction of block partials -> mean in d_out[0].
// ---------------------------------------------------------------------------
__global__ __launch_bounds__(TPB) void opening_loss_reduce_kernel(
    const float* __restrict__ partial, float* __restrict__ out, int n) {
  __shared__ float red[TPB];
  float acc = 0.0f;
  for (int i = threadIdx.x; i < n; i += TPB) acc += partial[i];
  red[threadIdx.x] = acc;
  __syncthreads();
  for (int s = TPB / 2; s > 0; s >>= 1) {
    if (threadIdx.x < s) red[threadIdx.x] += red[threadIdx.x + s];
    __syncthreads();
  }
  if (threadIdx.x == 0) out[0] = red[0] * INV_TOTAL;
}

// ---------------------------------------------------------------------------
extern "C" void kernel_launch(void* const* d_in, const int* in_sizes, int n_in,
                              void* d_out, int out_size, void* d_ws,
                              size_t ws_size, hipStream_t stream) {
  const float* labels = (const float*)d_in[0];
  float* partials = (float*)d_ws;            // 16384 floats = 64 KB scratch
  float* out = (float*)d_out;                // single f32 scalar

  opening_loss_tile_kernel<<<NBLOCKS, TPB, 0, stream>>>(labels, partials);
  opening_loss_reduce_kernel<<<1, TPB, 0, stream>>>(partials, out, NBLOCKS);
}